// BFPActivation_69758858822402
// MI455X (gfx1250) — compile-verified
//
#include <hip/hip_runtime.h>

// BFP quantization, block = channel vector (C=256) at each (n,h,w) of NCHW f32.
// Memory-bound: ~205MB read + ~205MB write -> ~17.6us roofline at 23.3 TB/s.
//
// Strategy: stage a [256 x 64] (channels x spatial) f32 tile into LDS with
// CDNA5 async global->LDS B128 copies (coalesced, VGPR-free), then each thread
// reduces its own channel column (max|x|), derives the shared exponent with
// frexp, and writes quantized values back with coalesced non-temporal stores.

#define N_    64
#define C_    256
#define HW_   3136              // 56*56
#define TILE  64                // spatial locations per workgroup (= blockDim.x)
#define BLOCKS_PER_N (HW_ / TILE)   // 49, exact -> a tile never straddles n

#if defined(__has_builtin)
#  if __has_builtin(__builtin_amdgcn_global_load_async_to_lds_b128)
#    define HAVE_ASYNC_LDS 1
#  endif
#  if __has_builtin(__builtin_amdgcn_s_wait_asynccnt)
#    define HAVE_WAIT_ASYNC 1
#  endif
#endif

// The async builtin takes pointers to 16-byte int vectors, AS-qualified.
typedef int v4i __attribute__((ext_vector_type(4)));
typedef __attribute__((address_space(1))) v4i gv4i;   // global
typedef __attribute__((address_space(3))) v4i lv4i;   // LDS

__global__ __launch_bounds__(TILE) void bfp_quant_kernel(
    const float* __restrict__ in, float* __restrict__ out)
{
    __shared__ float tile[C_ * TILE];   // 64 KB: [channel][spatial]

    const int tid = threadIdx.x;
    const int bx  = blockIdx.x;
    const int n   = bx / BLOCKS_PER_N;
    const int s0  = (bx - n * BLOCKS_PER_N) * TILE;

    const float* __restrict__ inb  = in  + (size_t)n * C_ * HW_ + s0;
    float*       __restrict__ outb = out + (size_t)n * C_ * HW_ + s0;

    // ---- Stage C_ x TILE floats into LDS, 16B per lane per op ----
    // Iteration i loads channels 4i..4i+3 x 64 spatial (1 KB per workgroup-op).
    {
        const int cg = tid >> 4;          // 0..3  channel subgroup
        const int sq = (tid & 15) << 2;   // 0..60 spatial quad base (16B aligned)
#if defined(HAVE_ASYNC_LDS)
        #pragma unroll 4
        for (int i = 0; i < C_ / 4; ++i) {
            const int c = 4 * i + cg;
            __builtin_amdgcn_global_load_async_to_lds_b128(
                (gv4i*)(inb + (size_t)c * HW_ + sq),
                (lv4i*)(&tile[c * TILE + sq]),
                /*imm offset=*/0, /*cpol=*/0);
        }
#  if defined(HAVE_WAIT_ASYNC)
        __builtin_amdgcn_s_wait_asynccnt(0);
#  else
        asm volatile("s_wait_asynccnt 0" ::: "memory");
#  endif
#else
        // Fallback: plain B128 load + LDS B128 store.
        #pragma unroll 4
        for (int i = 0; i < C_ / 4; ++i) {
            const int c = 4 * i + cg;
            const float4 v = *(const float4*)(inb + (size_t)c * HW_ + sq);
            *(float4*)(&tile[c * TILE + sq]) = v;
        }
#endif
    }
    __syncthreads();

    // ---- Per-thread block reduction: max|x| over 256 channels ----
    // Lane stride 4B in LDS -> conflict-free; 4 accumulators break the chain.
    float m0 = 0.0f, m1 = 0.0f, m2 = 0.0f, m3 = 0.0f;
    #pragma unroll 8
    for (int c = 0; c < C_; c += 4) {
        m0 = fmaxf(m0, fabsf(tile[(c + 0) * TILE + tid]));
        m1 = fmaxf(m1, fabsf(tile[(c + 1) * TILE + tid]));
        m2 = fmaxf(m2, fabsf(tile[(c + 2) * TILE + tid]));
        m3 = fmaxf(m3, fabsf(tile[(c + 3) * TILE + tid]));
    }
    const float maxab = fmaxf(fmaxf(m0, m1), fmaxf(m2, m3));

    // frexp convention: maxab = m * 2^e, m in [0.5, 1)
    int e;
    (void)frexpf(maxab, &e);
    const float delta     = ldexpf(1.0f, e - 3);   // 2^(e - mantissa_bits)
    const float inv_delta = ldexpf(1.0f, 3 - e);   // exact reciprocal (pow2)
    const bool  nz        = (maxab > 0.0f);

    // ---- Quantize + coalesced non-temporal stores ----
    #pragma unroll 4
    for (int c = 0; c < C_; ++c) {
        const float x = tile[c * TILE + tid];
        const float q = truncf(x * inv_delta) * delta;
        __builtin_nontemporal_store(nz ? q : 0.0f, &outb[(size_t)c * HW_ + tid]);
    }
}

extern "C" void kernel_launch(void* const* d_in, const int* in_sizes, int n_in,
                              void* d_out, int out_size, void* d_ws, size_t ws_size,
                              hipStream_t stream) {
    (void)in_sizes; (void)n_in; (void)out_size; (void)d_ws; (void)ws_size;
    const float* in  = (const float*)d_in[0];
    float*       out = (float*)d_out;
    const int grid = N_ * BLOCKS_PER_N;   // 64 * 49 = 3136 workgroups
    bfp_quant_kernel<<<grid, TILE, 0, stream>>>(in, out);
}